// Net_1967095021601
// MI455X (gfx1250) — compile-verified
//
#include <hip/hip_runtime.h>
#include <hip/hip_bf16.h>
#include <math.h>

typedef __attribute__((ext_vector_type(2))) float v2f;
typedef __attribute__((ext_vector_type(8))) float v8f;

#define NNODES 50000
#define NEDGES 800000
#define NB     512
#define DF     128

static inline int cdiv(long long a, long long b) { return (int)((a + b - 1) / b); }

// ---------------- utility kernels ----------------

__global__ void fill_zero_f32(float* __restrict__ p, long long n) {
  long long i = (long long)blockIdx.x * blockDim.x + threadIdx.x;
  if (i < n) p[i] = 0.0f;
}

__global__ void fill_zero_i32(int* __restrict__ p, long long n) {
  long long i = (long long)blockIdx.x * blockDim.x + threadIdx.x;
  if (i < n) p[i] = 0;
}

__global__ void copy_i32(const int* __restrict__ src, int* __restrict__ dst, long long n) {
  long long i = (long long)blockIdx.x * blockDim.x + threadIdx.x;
  if (i < n) dst[i] = src[i];
}

// in-degree histogram (int atomics; only E ops, done once for all 3 layers)
__global__ void degree_kernel(const int* __restrict__ ei, int* __restrict__ deg) {
  int e = blockIdx.x * blockDim.x + threadIdx.x;
  if (e < NEDGES) atomicAdd(&deg[ei[NEDGES + e]], 1);
}

// single-workgroup exclusive scan over deg[0..N) -> rowstart[0..N], LDS Hillis-Steele per chunk.
__global__ __launch_bounds__(1024) void scan_kernel(const int* __restrict__ deg,
                                                    int* __restrict__ rowstart) {
  __shared__ int lds[1024];
  __shared__ int carry;
  if (threadIdx.x == 0) carry = 0;
  __syncthreads();
  for (int base = 0; base < NNODES; base += 1024) {
    const int i = base + (int)threadIdx.x;
    const int v = (i < NNODES) ? deg[i] : 0;
    lds[threadIdx.x] = v;
    __syncthreads();
    for (int ofs = 1; ofs < 1024; ofs <<= 1) {
      const int t = (threadIdx.x >= (unsigned)ofs) ? lds[threadIdx.x - ofs] : 0;
      __syncthreads();
      lds[threadIdx.x] += t;
      __syncthreads();
    }
    const int incl = lds[threadIdx.x];
    if (i < NNODES) rowstart[i] = carry + incl - v;  // exclusive
    __syncthreads();
    if (threadIdx.x == 1023) carry += lds[1023];
    __syncthreads();
  }
  if (threadIdx.x == 0) rowstart[NNODES] = carry;
}

// bin edges by destination: csr_src[rowstart[dst] + k] = src
__global__ void csr_fill(const int* __restrict__ ei, int* __restrict__ cursor,
                         int* __restrict__ csr_src) {
  int e = blockIdx.x * blockDim.x + threadIdx.x;
  if (e < NEDGES) {
    const int dst = ei[NEDGES + e];
    const int pos = atomicAdd(&cursor[dst], 1);
    csr_src[pos] = ei[e];
  }
}

// batch is sorted; the first node of each graph is its root (segment_min).
__global__ void root_kernel(const int* __restrict__ batch, int* __restrict__ root) {
  int i = blockIdx.x * blockDim.x + threadIdx.x;
  if (i < NNODES) {
    if (i == 0 || batch[i] != batch[i - 1]) root[batch[i]] = i;
  }
}

// CSR gather-reduce: one wave per node, lane owns 4 contiguous features.
// Pure L2 reads (no float atomics, no zero-fill); mean fused into epilogue.
__global__ void aggregate_mean(const float* __restrict__ h, const int* __restrict__ rowstart,
                               const int* __restrict__ csr_src, float* __restrict__ mean) {
  long long gid = (long long)blockIdx.x * blockDim.x + threadIdx.x;
  long long node = gid >> 5;
  int lane = (int)(gid & 31);
  if (node >= NNODES) return;
  const int beg = rowstart[node];
  const int end = rowstart[node + 1];
  float ax = 0.f, ay = 0.f, az = 0.f, aw = 0.f;
  for (int j = beg; j < end; ++j) {
    const int src = csr_src[j];  // wave-uniform -> scalarizable
    const float4 v = *(const float4*)(h + (long long)src * DF + lane * 4);
    ax += v.x; ay += v.y; az += v.z; aw += v.w;
  }
  const float ic = 1.0f / fmaxf((float)(end - beg), 1.0f);
  float4 o = {ax * ic, ay * ic, az * ic, aw * ic};
  *(float4*)(mean + node * DF + lane * 4) = o;
}

// ---------------- fused SAGE layer: out = relu( mean @ Wl + h @ Wr + bl ) ----------------
// One wave -> one 16x16 fp32 output tile; 8 waves/block -> 16x128 tile. Two
// v_wmma_f32_16x16x4_f32 per K-step (64 per tile), pure load+WMMA inner loop.
__global__ __launch_bounds__(256) void sage_gemm_wmma(
    const float* __restrict__ mean, const float* __restrict__ h,
    const float* __restrict__ Wl, const float* __restrict__ Wr, const float* __restrict__ bl,
    float* __restrict__ out) {
  const int wave = threadIdx.x >> 5;  // 0..7 -> N tile
  const int lane = threadIdx.x & 31;
  const int half = lane >> 4;  // ISA A-16x4 layout: K pair split across lane halves
  const int lr   = lane & 15;
  const int m0   = blockIdx.x * 16;
  const int n0   = wave * 16;
  const long long rowA = (long long)(m0 + lr);

  v8f acc = {};
  for (int k0 = 0; k0 < DF; k0 += 4) {
    const int ka = k0 + 2 * half;
    v2f a, b;
    // neighborhood-mean path
    {
      const float2 t = *(const float2*)(mean + rowA * DF + ka);
      a.x = t.x; a.y = t.y;
    }
    b.x = Wl[(long long)ka * DF + n0 + lr];
    b.y = Wl[(long long)(ka + 1) * DF + n0 + lr];
    acc = __builtin_amdgcn_wmma_f32_16x16x4_f32(false, a, false, b, (short)0, acc, false, false);
    // self path
    {
      const float2 t = *(const float2*)(h + rowA * DF + ka);
      a.x = t.x; a.y = t.y;
    }
    b.x = Wr[(long long)ka * DF + n0 + lr];
    b.y = Wr[(long long)(ka + 1) * DF + n0 + lr];
    acc = __builtin_amdgcn_wmma_f32_16x16x4_f32(false, a, false, b, (short)0, acc, false, false);
  }

  // C/D layout: VGPR v, lanes 0-15 -> M=v, lanes 16-31 -> M=v+8; N = lane%16.
  const float bb = bl[n0 + lr];
#pragma unroll
  for (int v = 0; v < 8; ++v) {
    const int m = m0 + v + 8 * half;
    float val = acc[v] + bb;
    out[(long long)m * DF + n0 + lr] = fmaxf(val, 0.0f);
  }
}

// ---------------- generic GEMM + bias + relu (M,K,N multiples of 16; K mult of 4) ----------------
__global__ __launch_bounds__(256) void gemm_bias_relu_wmma(
    const float* __restrict__ A, const float* __restrict__ W, const float* __restrict__ bias,
    float* __restrict__ out, int M, int K, int Nout) {
  const int ntiles = Nout >> 4;
  const int tile = blockIdx.x * 8 + (threadIdx.x >> 5);
  const int mt = tile / ntiles;
  const int nt = tile - mt * ntiles;
  if (mt * 16 >= M) return;  // wave-uniform -> surviving waves keep EXEC all-ones for WMMA

  const int lane = threadIdx.x & 31;
  const int half = lane >> 4;
  const int lr   = lane & 15;
  const int m0   = mt * 16;
  const int n0   = nt * 16;
  const long long rowA = (long long)(m0 + lr);

  v8f acc = {};
  for (int k0 = 0; k0 < K; k0 += 4) {
    const int ka = k0 + 2 * half;
    v2f a, b;
    const float2 t = *(const float2*)(A + rowA * K + ka);
    a.x = t.x; a.y = t.y;
    b.x = W[(long long)ka * Nout + n0 + lr];
    b.y = W[(long long)(ka + 1) * Nout + n0 + lr];
    acc = __builtin_amdgcn_wmma_f32_16x16x4_f32(false, a, false, b, (short)0, acc, false, false);
  }
  const float bb = bias[n0 + lr];
#pragma unroll
  for (int v = 0; v < 8; ++v) {
    const int m = m0 + v + 8 * half;
    float val = acc[v] + bb;
    out[(long long)m * Nout + n0 + lr] = fmaxf(val, 0.0f);
  }
}

// ---------------- global max pool (post-ReLU >= 0 -> int compare == float compare) ----------------
__global__ void pool_max_kernel(const float* __restrict__ h, const int* __restrict__ batch,
                                int* __restrict__ pooled) {
  long long gid = (long long)blockIdx.x * blockDim.x + threadIdx.x;
  long long node = gid >> 5;
  int lane = (int)(gid & 31);
  if (node >= NNODES) return;
  int b = batch[node];
  const float4 v = *(const float4*)(h + node * DF + lane * 4);
  int* out = pooled + (long long)b * DF + lane * 4;
  atomicMax(out + 0, __float_as_int(v.x));
  atomicMax(out + 1, __float_as_int(v.y));
  atomicMax(out + 2, __float_as_int(v.z));
  atomicMax(out + 3, __float_as_int(v.w));
}

// ---------------- head: 128->2 x2, concat, 4->1, sigmoid ----------------
__global__ void head_kernel(const float* __restrict__ f2, const float* __restrict__ x,
                            const int* __restrict__ root,
                            const float* __restrict__ Wsm, const float* __restrict__ bsm,
                            const float* __restrict__ Wnews, const float* __restrict__ bnews,
                            const float* __restrict__ Wcat, const float* __restrict__ bcat,
                            float* __restrict__ outp) {
  int b = blockIdx.x * blockDim.x + threadIdx.x;
  if (b >= NB) return;
  float sm0 = bsm[0], sm1 = bsm[1];
  const float* row = f2 + (long long)b * DF;
  for (int k = 0; k < DF; ++k) {
    float v = row[k];
    sm0 += v * Wsm[k * 2 + 0];
    sm1 += v * Wsm[k * 2 + 1];
  }
  sm0 = fmaxf(sm0, 0.0f);
  sm1 = fmaxf(sm1, 0.0f);
  const float* xr = x + (long long)root[b] * DF;
  float nw0 = bnews[0], nw1 = bnews[1];
  for (int k = 0; k < DF; ++k) {
    float v = xr[k];
    nw0 += v * Wnews[k * 2 + 0];
    nw1 += v * Wnews[k * 2 + 1];
  }
  nw0 = fmaxf(nw0, 0.0f);
  nw1 = fmaxf(nw1, 0.0f);
  float z = bcat[0] + sm0 * Wcat[0] + sm1 * Wcat[1] + nw0 * Wcat[2] + nw1 * Wcat[3];
  outp[b] = 1.0f / (1.0f + __expf(-z));
}

// ---------------- launcher ----------------

extern "C" void kernel_launch(void* const* d_in, const int* in_sizes, int n_in,
                              void* d_out, int out_size, void* d_ws, size_t ws_size,
                              hipStream_t stream) {
  const float* x      = (const float*)d_in[0];
  const int*   ei     = (const int*)d_in[1];
  const int*   batch  = (const int*)d_in[2];
  const float* Wl1 = (const float*)d_in[3],  *Wr1 = (const float*)d_in[4],  *bl1 = (const float*)d_in[5];
  const float* Wl2 = (const float*)d_in[6],  *Wr2 = (const float*)d_in[7],  *bl2 = (const float*)d_in[8];
  const float* Wl3 = (const float*)d_in[9],  *Wr3 = (const float*)d_in[10], *bl3 = (const float*)d_in[11];
  const float* Wf1 = (const float*)d_in[12], *bf1 = (const float*)d_in[13];
  const float* Wf2 = (const float*)d_in[14], *bf2 = (const float*)d_in[15];
  const float* Wsm = (const float*)d_in[16], *bsm = (const float*)d_in[17];
  const float* Wnw = (const float*)d_in[18], *bnw = (const float*)d_in[19];
  const float* Wct = (const float*)d_in[20], *bct = (const float*)d_in[21];
  float* outp = (float*)d_out;

  // workspace carve-up (256B aligned)
  char* ws = (char*)d_ws;
  size_t off = 0;
  auto take = [&](size_t bytes) {
    char* p = ws + off;
    off = (off + bytes + 255) & ~(size_t)255;
    return p;
  };
  float* mean    = (float*)take((size_t)NNODES * DF * 4);
  float* hA      = (float*)take((size_t)NNODES * DF * 4);
  float* hB      = (float*)take((size_t)NNODES * DF * 4);
  int*   deg     = (int*)take((size_t)NNODES * 4);
  int*   rowstart= (int*)take((size_t)(NNODES + 1) * 4);
  int*   cursor  = (int*)take((size_t)NNODES * 4);
  int*   csr_src = (int*)take((size_t)NEDGES * 4);
  float* pooled  = (float*)take((size_t)NB * DF * 4);
  float* f1o     = (float*)take((size_t)NB * 256 * 4);
  float* f2o     = (float*)take((size_t)NB * DF * 4);
  int*   root    = (int*)take((size_t)NB * 4);
  (void)ws_size; (void)n_in; (void)in_sizes; (void)out_size;

  const int nodeWaveBlocks = cdiv((long long)NNODES * 32, 256);

  // --- build CSR (once; shared by all 3 layers) + roots ---
  fill_zero_i32<<<cdiv(NNODES, 256), 256, 0, stream>>>(deg, NNODES);
  degree_kernel<<<cdiv(NEDGES, 256), 256, 0, stream>>>(ei, deg);
  scan_kernel<<<1, 1024, 0, stream>>>(deg, rowstart);
  copy_i32<<<cdiv(NNODES, 256), 256, 0, stream>>>(rowstart, cursor, NNODES);
  csr_fill<<<cdiv(NEDGES, 256), 256, 0, stream>>>(ei, cursor, csr_src);
  root_kernel<<<cdiv(NNODES, 256), 256, 0, stream>>>(batch, root);

  // --- layer 1: x -> hA ---
  aggregate_mean<<<nodeWaveBlocks, 256, 0, stream>>>(x, rowstart, csr_src, mean);
  sage_gemm_wmma<<<NNODES / 16, 256, 0, stream>>>(mean, x, Wl1, Wr1, bl1, hA);
  // --- layer 2: hA -> hB ---
  aggregate_mean<<<nodeWaveBlocks, 256, 0, stream>>>(hA, rowstart, csr_src, mean);
  sage_gemm_wmma<<<NNODES / 16, 256, 0, stream>>>(mean, hA, Wl2, Wr2, bl2, hB);
  // --- layer 3: hB -> hA ---
  aggregate_mean<<<nodeWaveBlocks, 256, 0, stream>>>(hB, rowstart, csr_src, mean);
  sage_gemm_wmma<<<NNODES / 16, 256, 0, stream>>>(mean, hB, Wl3, Wr3, bl3, hA);

  // --- global max pool (init 0 correct: post-ReLU features >= 0) ---
  fill_zero_f32<<<cdiv((long long)NB * DF, 256), 256, 0, stream>>>(pooled, (long long)NB * DF);
  pool_max_kernel<<<nodeWaveBlocks, 256, 0, stream>>>(hA, batch, (int*)pooled);

  // --- MLP (WMMA) + head ---
  gemm_bias_relu_wmma<<<(NB / 16) * (256 / 16) / 8, 256, 0, stream>>>(pooled, Wf1, bf1, f1o, NB, 128, 256);
  gemm_bias_relu_wmma<<<(NB / 16) * (128 / 16) / 8, 256, 0, stream>>>(f1o, Wf2, bf2, f2o, NB, 256, 128);
  head_kernel<<<cdiv(NB, 256), 256, 0, stream>>>(f2o, x, root, Wsm, bsm, Wnw, bnw, Wct, bct, outp);
}